// LearnableTD_19971597926883
// MI455X (gfx1250) — compile-verified
//
#include <hip/hip_runtime.h>
#include <hip/hip_bf16.h>

// TD(lambda) backward scan, B=65536 rows x S=512 steps, fp32.
// Memory-bound streaming kernel (~515 MB => ~22us at 23.3 TB/s).
// Strategy: 1 thread per row, S tiled in chunks of 16, tiles staged
// through LDS with CDNA5 async global<->LDS copies for full coalescing.

#define S_LEN   512
#define SP1     513           // values row length (S+1)
#define TS      16            // steps per chunk
#define ROWS    128           // rows per block (== blockDim.x)
#define BLOCK   128
#define NCHUNK  (S_LEN / TS)  // 32
#define LSTR    20            // LDS row stride in floats (16 data + 4 pad, 80B: 16B aligned)

// ---------------- CDNA5 async data movement helpers ----------------

// Builtin signature (from hipcc diagnostic): pointers to 16B int vectors,
// global (AS1) first, LDS (AS3) second, then imm offset + imm cpol.
typedef int v4i __attribute__((vector_size(16)));
typedef __attribute__((address_space(1))) v4i* gv4i_p;
typedef __attribute__((address_space(3))) v4i* lv4i_p;

__device__ __forceinline__ void wait_async0() {
#if __has_builtin(__builtin_amdgcn_s_wait_asynccnt)
    __builtin_amdgcn_s_wait_asynccnt(0);
#else
    asm volatile("s_wait_asynccnt 0" ::: "memory");
#endif
}

// 16B global -> LDS, tracked by ASYNCcnt (falls back to reg copy if builtin absent)
__device__ __forceinline__ void g2l16(const float* g, float* l) {
#if __has_builtin(__builtin_amdgcn_global_load_async_to_lds_b128)
    __builtin_amdgcn_global_load_async_to_lds_b128(
        (gv4i_p)g, (lv4i_p)l, 0 /*offset*/, 0 /*cpol*/);
#else
    *(float4*)l = *(const float4*)g;
#endif
}

// 16B LDS -> global, tracked by ASYNCcnt
__device__ __forceinline__ void l2g16(float* g, const float* l) {
#if __has_builtin(__builtin_amdgcn_global_store_async_from_lds_b128)
    __builtin_amdgcn_global_store_async_from_lds_b128(
        (gv4i_p)g, (lv4i_p)l, 0 /*offset*/, 0 /*cpol*/);
#else
    *(float4*)g = *(const float4*)l;
#endif
}

// -------------------------------------------------------------------

__global__ __launch_bounds__(BLOCK)
void td_lambda_scan(const float* __restrict__ raw_gamma,
                    const float* __restrict__ raw_lambd,
                    const float* __restrict__ values,   // (B, S+1)
                    const float* __restrict__ rewards,  // (B, S)
                    const float* __restrict__ dones,    // (B, S)
                    float* __restrict__ out)            // (B, S)
{
    __shared__ __align__(16) float sh_lam[S_LEN];
    __shared__ __align__(16) float sh_r[ROWS * LSTR];
    __shared__ __align__(16) float sh_d[ROWS * LSTR];
    __shared__ __align__(16) float sh_v[ROWS * LSTR];
    __shared__ __align__(16) float sh_o[ROWS * LSTR];

    const int t  = threadIdx.x;
    const int b0 = blockIdx.x * ROWS;

    // Stage lambda(s) = max(tanh(raw_lambd[s]), 1e-8) once per block.
    for (int s = t; s < S_LEN; s += BLOCK)
        sh_lam[s] = fmaxf(tanhf(raw_lambd[s]), 1e-8f);
    const float gamma = fmaxf(tanhf(raw_gamma[0]), 1e-8f);

    // This thread owns batch row (b0 + t).
    const int    row = t;
    const size_t b   = (size_t)(b0 + row);
    float carry = values[b * SP1 + S_LEN];   // bootstrap: values[:, -1]

    // Cooperative tile-copy pattern: 4 lanes per row (4x b128 = 64B/row),
    // 32 rows per pass, 4 passes -> lane-contiguous global addresses.
    const int lrow = t >> 2;         // 0..31
    const int lcol = (t & 3) << 2;   // 0,4,8,12

    for (int c = NCHUNK - 1; c >= 0; --c) {
        const int sb = c * TS;

        // ---- async load r/d/v tiles for this chunk (coalesced) ----
        #pragma unroll
        for (int p = 0; p < ROWS / 32; ++p) {
            const int    rr = lrow + p * 32;
            const size_t gb = (size_t)(b0 + rr);
            g2l16(rewards + gb * S_LEN + sb + lcol,     &sh_r[rr * LSTR + lcol]);
            g2l16(dones   + gb * S_LEN + sb + lcol,     &sh_d[rr * LSTR + lcol]);
            g2l16(values  + gb * SP1   + sb + 1 + lcol, &sh_v[rr * LSTR + lcol]);
        }
        // Each wave drains its own async ops (loads for this chunk AND its
        // output stores from the previous chunk), then block-wide barrier:
        // after this point all LDS tiles (incl. sh_o) are free/ready.
        wait_async0();
        __syncthreads();

        // ---- pull my 16-step slice into registers (b128 LDS reads) ----
        float rv[TS], dv[TS], vv[TS], lv[TS], ov[TS];
        {
            const float4* r4 = (const float4*)&sh_r[row * LSTR];
            const float4* d4 = (const float4*)&sh_d[row * LSTR];
            const float4* v4 = (const float4*)&sh_v[row * LSTR];
            const float4* l4 = (const float4*)&sh_lam[sb];
            #pragma unroll
            for (int q = 0; q < TS / 4; ++q) {
                float4 x;
                x = r4[q]; rv[4*q]=x.x; rv[4*q+1]=x.y; rv[4*q+2]=x.z; rv[4*q+3]=x.w;
                x = d4[q]; dv[4*q]=x.x; dv[4*q+1]=x.y; dv[4*q+2]=x.z; dv[4*q+3]=x.w;
                x = v4[q]; vv[4*q]=x.x; vv[4*q+1]=x.y; vv[4*q+2]=x.z; vv[4*q+3]=x.w;
                x = l4[q]; lv[4*q]=x.x; lv[4*q+1]=x.y; lv[4*q+2]=x.z; lv[4*q+3]=x.w;
            }
        }

        // ---- sequential scan, back to front, fully unrolled in regs ----
        #pragma unroll
        for (int j = TS - 1; j >= 0; --j) {
            const float a   = gamma * (1.0f - dv[j]);
            const float mix = (1.0f - lv[j]) * vv[j] + lv[j] * carry;
            const float ret = fmaf(a, mix, rv[j]);
            ov[j] = ret;
            carry = ret;
        }

        // ---- write my results to the LDS output tile ----
        {
            float4* o4 = (float4*)&sh_o[row * LSTR];
            #pragma unroll
            for (int q = 0; q < TS / 4; ++q)
                o4[q] = make_float4(ov[4*q], ov[4*q+1], ov[4*q+2], ov[4*q+3]);
        }
        __syncthreads();   // sh_o fully written before cooperative store

        // ---- async store output tile LDS -> global (coalesced) ----
        #pragma unroll
        for (int p = 0; p < ROWS / 32; ++p) {
            const int    rr = lrow + p * 32;
            const size_t gb = (size_t)(b0 + rr);
            l2g16(out + gb * S_LEN + sb + lcol, &sh_o[rr * LSTR + lcol]);
        }
        // Drained by wait_async0() at the top of the next iteration;
        // s_endpgm performs an implicit wait-idle for the final chunk.
    }
}

extern "C" void kernel_launch(void* const* d_in, const int* in_sizes, int n_in,
                              void* d_out, int out_size, void* d_ws, size_t ws_size,
                              hipStream_t stream) {
    const float* raw_gamma = (const float*)d_in[0];   // (1,)
    const float* raw_lambd = (const float*)d_in[1];   // (S,)
    const float* values    = (const float*)d_in[2];   // (B, S+1)
    const float* rewards   = (const float*)d_in[3];   // (B, S)
    const float* dones     = (const float*)d_in[4];   // (B, S)
    float*       out       = (float*)d_out;           // (B, S)

    const int B = in_sizes[3] / S_LEN;                // rewards flat count = B*S
    dim3 grid(B / ROWS), block(BLOCK);
    td_lambda_scan<<<grid, block, 0, stream>>>(raw_gamma, raw_lambd, values,
                                               rewards, dones, out);
}